// QuantumQCNN_65481071408012
// MI455X (gfx1250) — compile-verified
//
#include <hip/hip_runtime.h>
#include <hip/hip_bf16.h>

// ============================================================================
// QuantumQCNN for MI455X (gfx1250, wave32).
//
// Math reduction: output = sigmoid(Z @ W.T + b) with
//   Z_j = prod_{m in {j, j-2, j-4, ...}} cos(x_m)
// RZ phases cancel in |amp|^2; the doubled CNOT chain is a GF(2)-linear basis
// permutation whose Z-signature is the even-lag prefix set S_j.
//
// Bandwidth-bound: 4MB in + 2MB out (~0.26us at 23.3 TB/s). One pass:
// float4 loads, v_cos_f32, even/odd prefix products, 8->4 affine map via two
// chained V_WMMA_F32_16X16X4_F32 (bias pre-loaded into the accumulator),
// sigmoid via v_exp_f32 + v_rcp_f32 (fast HW reciprocal, ~1 ULP).
// ============================================================================

typedef float v2f __attribute__((ext_vector_type(2)));
typedef float v8f __attribute__((ext_vector_type(8)));

__global__ __launch_bounds__(256) void qcnn_kernel(
    const float* __restrict__ x,     // [B, 8]
    const float* __restrict__ W,     // [4, 8]
    const float* __restrict__ bias,  // [4]
    float* __restrict__ out,         // [B, 4]
    int B)
{
    // Per-thread Z staging: 256 threads * 8 floats = 8 KB LDS (WGP has 320 KB).
    __shared__ float zbuf[256 * 8];

    const int tid  = threadIdx.x;
    const int lane = tid & 31;
    const int wave = tid >> 5;
    const int half = lane >> 4;   // 0: lanes 0-15, 1: lanes 16-31
    const int l16  = lane & 15;

    const int  blk_base = blockIdx.x * 256;
    const bool tail     = (blk_base + 256 > B);   // never true when B%256==0

    int sample = blk_base + tid;
    int s_ld   = sample < B ? sample : B - 1;     // clamp load for tail safety

    // ---- 1. coalesced load of the 8 encoding angles (32B per lane) ----
    const float4* xp = (const float4*)(x + (size_t)s_ld * 8);
    float4 lo = xp[0];
    float4 hi = xp[1];

    // ---- 2. cos() per wire, even/odd chained prefix products -> Z[8] ----
    float c0 = __cosf(lo.x), c1 = __cosf(lo.y), c2 = __cosf(lo.z), c3 = __cosf(lo.w);
    float c4 = __cosf(hi.x), c5 = __cosf(hi.y), c6 = __cosf(hi.z), c7 = __cosf(hi.w);

    float z[8];
    z[0] = c0;        z[1] = c1;
    z[2] = c2 * z[0]; z[3] = c3 * z[1];
    z[4] = c4 * z[2]; z[5] = c5 * z[3];
    z[6] = c6 * z[4]; z[7] = c7 * z[5];

#pragma unroll
    for (int k = 0; k < 8; ++k) zbuf[tid * 8 + k] = z[k];
    __syncthreads();

    // ---- 3. B-matrix fragments: B[k][n] = W[n][k], n<4 live, rest zero.
    // 4x16 f32 layout mirrors the ISA 16x4 A layout:
    //   VGPR0 = row K=2*half, VGPR1 = row K=2*half+1, lane -> column N.
    v2f bf0 = {0.f, 0.f}, bf1 = {0.f, 0.f};
    float bn = 0.f;
    if (l16 < 4) {
        const float* wr = W + l16 * 8;          // W row n: W[n][0..7]
        bf0.x = wr[2 * half + 0];               // K = 2*half      (step 0)
        bf0.y = wr[2 * half + 1];               // K = 2*half + 1
        bf1.x = wr[4 + 2 * half + 0];           // K = 4 + 2*half  (step 1)
        bf1.y = wr[4 + 2 * half + 1];
        bn = bias[l16];
    }

    const float* zw = &zbuf[wave * 32 * 8];     // this wave's 32 samples

    // ---- 4. two 16-sample M-tiles per wave, 2 chained WMMAs each (K=8) ----
#pragma unroll
    for (int t = 0; t < 2; ++t) {
        // A fragment, 16x4 f32 ISA layout: lane -> row M=l16,
        // VGPR0 = K=2*half, VGPR1 = K=2*half+1.
        const float* zs = zw + (t * 16 + l16) * 8;
        v2f a0, a1;
        a0.x = zs[2 * half + 0];
        a0.y = zs[2 * half + 1];
        a1.x = zs[4 + 2 * half + 0];
        a1.y = zs[4 + 2 * half + 1];

        // Accumulator pre-loaded with bias (C/D layout: N = l16 everywhere).
        v8f acc;
#pragma unroll
        for (int r = 0; r < 8; ++r) acc[r] = bn;

        // D = A(16x4) x B(4x16) + C, twice to cover K=8. Full EXEC here.
        acc = __builtin_amdgcn_wmma_f32_16x16x4_f32(
                  false, a0, false, bf0, (short)0, acc, false, false);
        acc = __builtin_amdgcn_wmma_f32_16x16x4_f32(
                  false, a1, false, bf1, (short)0, acc, false, false);

        // ---- 5. sigmoid + store. D layout: VGPR r, lanes 0-15 -> M=r,
        // lanes 16-31 -> M=r+8; column N = l16 (only N<4 is real output).
        if (l16 < 4) {
            int m0 = blk_base + wave * 32 + t * 16 + half * 8;
            if (!tail) {
                // hot path: no per-element guards, no branches
#pragma unroll
                for (int r = 0; r < 8; ++r) {
                    float sig = __builtin_amdgcn_rcpf(1.0f + __expf(-acc[r]));
                    out[(size_t)(m0 + r) * 4 + l16] = sig;
                }
            } else {
#pragma unroll
                for (int r = 0; r < 8; ++r) {
                    if (m0 + r < B) {
                        float sig = __builtin_amdgcn_rcpf(1.0f + __expf(-acc[r]));
                        out[(size_t)(m0 + r) * 4 + l16] = sig;
                    }
                }
            }
        }
    }
}

extern "C" void kernel_launch(void* const* d_in, const int* in_sizes, int n_in,
                              void* d_out, int out_size, void* d_ws, size_t ws_size,
                              hipStream_t stream) {
    // setup_inputs order: inputs [B,8], rz_params [2,8] (provably unused:
    // diagonal phases cancel in |amp|^2), W [4,8], b [4].
    const float* x    = (const float*)d_in[0];
    const float* W    = (const float*)d_in[2];
    const float* bias = (const float*)d_in[3];
    float* out        = (float*)d_out;

    int B = in_sizes[0] / 8;
    int grid = (B + 255) / 256;
    qcnn_kernel<<<grid, 256, 0, stream>>>(x, W, bias, out, B);
}